// CRFOutputLayer_66408784331627
// MI455X (gfx1250) — compile-verified
//
#include <hip/hip_runtime.h>

typedef __attribute__((ext_vector_type(2))) float v2f;
typedef __attribute__((ext_vector_type(8))) float v8f;

#define BB   128
#define TT   1024
#define HH   512
#define KK   50
#define KP   64          // padded tag dimension
#define MM   (BB * TT)   // 131072 rows in the projection GEMM

// Workspace layout (bytes):
//   feats : [MM][KP] f32            = 33,554,432
//   Wpad  : [KP][HH] f32 (rows >=50 zero) = 131,072
#define FEATS_WS_BYTES ((size_t)MM * KP * 4)

// ---------------------------------------------------------------------------
// Kernel 0: zero-pad W (50x512) into Wpad (64x512) so the GEMM B-loads need
// no bounds check (kills the per-iteration EXEC save/restore in the hot loop).
// ---------------------------------------------------------------------------
__global__ __launch_bounds__(256) void crf_pad_w(
    const float* __restrict__ W,      // [KK, HH]
    float* __restrict__ Wpad)         // [KP, HH]
{
    const int idx = blockIdx.x * 256 + threadIdx.x;   // grid covers KP*HH
    const int n = idx >> 9;           // /HH
    if (idx < KP * HH)
        Wpad[idx] = (n < KK) ? W[idx] : 0.0f;
}

// ---------------------------------------------------------------------------
// Kernel 1: feats[m, k] = sum_h hidden[m, h] * W[k, h] + b[k]
// fp32 WMMA 16x16x4. One wave owns TWO 16-row M-tiles and all 4 N-tiles:
// per k-step that's 2 A loads + 4 B loads feeding 8 WMMAs (B reuse).
// A layout (32-bit A 16x4): lanes 0-15 hold {K=0,K=1}, lanes 16-31 hold
// {K=2,K=3} for row M=lane&15; B mirrored.
// ---------------------------------------------------------------------------
__global__ __launch_bounds__(256) void crf_gemm_wmma(
    const float* __restrict__ hidden,   // [MM, HH]
    const float* __restrict__ Wpad,     // [KP, HH]
    const float* __restrict__ bias,     // [KK]
    float* __restrict__ feats)          // [MM, KP]
{
    const int wave  = threadIdx.x >> 5;
    const int lane  = threadIdx.x & 31;
    const int half  = lane >> 4;              // 0: K=0,1   1: K=2,3
    const int l15   = lane & 15;
    // Each wave: tiles (2*gw) and (2*gw+1) -> rows [mbase, mbase+32)
    const int mbase = (blockIdx.x * 8 + wave) * 32;

    const float* arow0 = hidden + (size_t)(mbase + l15) * HH + half * 2;
    const float* arow1 = arow0 + (size_t)16 * HH;
    const float* brow  = Wpad + (size_t)l15 * HH + half * 2;

    v8f accA0 = {}, accA1 = {}, accA2 = {}, accA3 = {};
    v8f accB0 = {}, accB1 = {}, accB2 = {}, accB3 = {};

    for (int h = 0; h < HH; h += 4) {
        v2f a0 = *(const v2f*)(arow0 + h);
        v2f a1 = *(const v2f*)(arow1 + h);

        v2f b0 = *(const v2f*)(brow + h);                     // n = l15
        v2f b1 = *(const v2f*)(brow + 16 * HH + h);           // n = 16 + l15
        v2f b2 = *(const v2f*)(brow + 32 * HH + h);           // n = 32 + l15
        v2f b3 = *(const v2f*)(brow + 48 * HH + h);           // n = 48 + l15 (padded)

        accA0 = __builtin_amdgcn_wmma_f32_16x16x4_f32(false, a0, false, b0,
                                                      (short)0, accA0, false, false);
        accB0 = __builtin_amdgcn_wmma_f32_16x16x4_f32(false, a1, false, b0,
                                                      (short)0, accB0, false, false);
        accA1 = __builtin_amdgcn_wmma_f32_16x16x4_f32(false, a0, false, b1,
                                                      (short)0, accA1, false, false);
        accB1 = __builtin_amdgcn_wmma_f32_16x16x4_f32(false, a1, false, b1,
                                                      (short)0, accB1, false, false);
        accA2 = __builtin_amdgcn_wmma_f32_16x16x4_f32(false, a0, false, b2,
                                                      (short)0, accA2, false, false);
        accB2 = __builtin_amdgcn_wmma_f32_16x16x4_f32(false, a1, false, b2,
                                                      (short)0, accB2, false, false);
        accA3 = __builtin_amdgcn_wmma_f32_16x16x4_f32(false, a0, false, b3,
                                                      (short)0, accA3, false, false);
        accB3 = __builtin_amdgcn_wmma_f32_16x16x4_f32(false, a1, false, b3,
                                                      (short)0, accB3, false, false);
    }

    // C/D layout: VGPR r, lanes 0-15 -> M = r, lanes 16-31 -> M = r + 8; N = l15.
    v8f accsA[4] = {accA0, accA1, accA2, accA3};
    v8f accsB[4] = {accB0, accB1, accB2, accB3};
#pragma unroll
    for (int c = 0; c < 4; ++c) {
        const int n = c * 16 + l15;
        const float bval = (n < KK) ? bias[n] : 0.0f;
#pragma unroll
        for (int r = 0; r < 8; ++r) {
            const int m = mbase + r + half * 8;
            feats[(size_t)m * KP + n]        = accsA[c][r] + bval;
            feats[(size_t)(m + 16) * KP + n] = accsB[c][r] + bval;
        }
    }
}

// ---------------------------------------------------------------------------
// Kernel 2: Viterbi DP + backtrack. One workgroup (256 threads) per batch
// element. Backpointers, transitions and the DP vector all stay in LDS; the
// backtrack is an LDS-only pointer chase. No inter-workgroup sync needed.
//
// Thread layout: j = tid & 63 (tag, active j < 50), part = tid >> 6 (0..3),
// each part reduces a 13-wide slice of the "from" index i, combined by part 0.
// ---------------------------------------------------------------------------
#define TRANS_OFF 0                       // 50*50 f32 = 10000 B
#define VBUF_OFF  10000                   // 2*64  f32 =   512 B
#define PV_OFF    10512                   // 4*64  f32 =  1024 B
#define PIDX_OFF  11536                   // 4*64  i32 =  1024 B
#define TAGS_OFF  12560                   // 1024  u8  =  1024 B
#define BP_OFF    13584                   // 1024*50 u8 = 51200 B
#define SMEM_SIZE 64784

__global__ __launch_bounds__(256) void crf_viterbi(
    const float* __restrict__ feats,      // [MM, KP]
    const float* __restrict__ trans,      // [KK, KK]  (from i -> to j)
    const float* __restrict__ start_t,    // [KK]
    const float* __restrict__ stop_t,     // [KK]
    float* __restrict__ out)              // [BB] scores, then [BB*TT] tags
{
    extern __shared__ char smem[];
    float*         trans_s = (float*)(smem + TRANS_OFF);   // [50][50]
    float*         vbuf    = (float*)(smem + VBUF_OFF);    // [2][64]
    float*         pv      = (float*)(smem + PV_OFF);      // [4][64]
    int*           pidx    = (int*)  (smem + PIDX_OFF);    // [4][64]
    unsigned char* tags_s  = (unsigned char*)(smem + TAGS_OFF); // [1024]
    unsigned char* bp_s    = (unsigned char*)(smem + BP_OFF);   // [1024][50]

    const int b    = blockIdx.x;
    const int tid  = threadIdx.x;
    const int j    = tid & 63;
    const int part = tid >> 6;

    const float* fb = feats + (size_t)b * TT * KP;

    // Stage transitions in LDS.
    for (int idx = tid; idx < KK * KK; idx += 256)
        trans_s[idx] = trans[idx];

    if (tid < KP)
        vbuf[tid] = (tid < KK) ? (fb[tid] + start_t[tid]) : -3.0e38f;
    __syncthreads();

    const int i0 = part * 13;
    const int i1 = (i0 + 13 < KK) ? (i0 + 13) : KK;
    int cur = 0;

    for (int t = 1; t < TT; ++t) {
        float best = -3.0e38f;
        int   bi   = 0;
        if (j < KK) {
            const float* vr = vbuf + cur * KP;
            for (int i = i0; i < i1; ++i) {
                const float s = vr[i] + trans_s[i * KK + j];
                if (s > best) { best = s; bi = i; }   // keeps first max (ascending i)
            }
        }
        pv[part * 64 + j]   = best;
        pidx[part * 64 + j] = bi;
        __syncthreads();

        if (part == 0 && j < KK) {
            float bb = pv[j];
            int   ii = pidx[j];
#pragma unroll
            for (int p = 1; p < 4; ++p) {
                const float s = pv[p * 64 + j];
                if (s > bb) { bb = s; ii = pidx[p * 64 + j]; }
            }
            bp_s[t * KK + j] = (unsigned char)ii;
            vbuf[(cur ^ 1) * KP + j] = fb[(size_t)t * KP + j] + bb;
        }
        __syncthreads();
        cur ^= 1;
    }

    // Terminal argmax + LDS-only backtrack (thread 0, sequential but cheap).
    if (tid == 0) {
        float bb = -3.0e38f;
        int   bt = 0;
        for (int k = 0; k < KK; ++k) {
            const float s = vbuf[cur * KP + k] + stop_t[k];
            if (s > bb) { bb = s; bt = k; }
        }
        out[b] = bb;
        int tag = bt;
        tags_s[TT - 1] = (unsigned char)tag;
        for (int t = TT - 1; t >= 1; --t) {
            tag = bp_s[t * KK + tag];
            tags_s[t - 1] = (unsigned char)tag;
        }
    }
    __syncthreads();

    // Coalesced tag writeback (int values emitted as floats per output dtype).
    float* tags_out = out + BB + (size_t)b * TT;
    for (int t = tid; t < TT; t += 256)
        tags_out[t] = (float)tags_s[t];
}

extern "C" void kernel_launch(void* const* d_in, const int* in_sizes, int n_in,
                              void* d_out, int out_size, void* d_ws, size_t ws_size,
                              hipStream_t stream) {
    (void)in_sizes; (void)n_in; (void)out_size; (void)ws_size;

    const float* hidden  = (const float*)d_in[0];
    const float* W       = (const float*)d_in[1];
    const float* bias    = (const float*)d_in[2];
    const float* trans   = (const float*)d_in[3];
    const float* start_t = (const float*)d_in[4];
    const float* stop_t  = (const float*)d_in[5];
    float*       out     = (float*)d_out;

    float* feats = (float*)d_ws;
    float* Wpad  = (float*)((char*)d_ws + FEATS_WS_BYTES);

    // Prologue: zero-pad W into [KP][HH].
    crf_pad_w<<<dim3((KP * HH + 255) / 256), dim3(256), 0, stream>>>(W, Wpad);

    // GEMM: 8192 M-tiles, 2 tiles per wave, 8 waves per 256-thread block.
    crf_gemm_wmma<<<dim3(MM / 16 / 2 / 8), dim3(256), 0, stream>>>(hidden, Wpad,
                                                                   bias, feats);

    // Viterbi: one workgroup per batch element; all bp state in LDS.
    crf_viterbi<<<dim3(BB), dim3(256), SMEM_SIZE, stream>>>(feats, trans, start_t,
                                                            stop_t, out);
}